// L2GENConv_84859963834442
// MI455X (gfx1250) — compile-verified
//
#include <hip/hip_runtime.h>
#include <hip/hip_bf16.h>
#include <math.h>

#define NNODES 50000
#define NEDGES 800000
#define DIN    128
#define DHID   200
#define MSG_EPS 1e-7f
#define BN_EPS  1e-5f

typedef float v2f __attribute__((ext_vector_type(2)));
typedef float v8f __attribute__((ext_vector_type(8)));

// ---------------- init: zero m / s / wsum ----------------
__global__ void init3_kernel(float* __restrict__ m, float* __restrict__ s,
                             float* __restrict__ w, int total) {
    for (int i = blockIdx.x * blockDim.x + threadIdx.x; i < total;
         i += gridDim.x * blockDim.x) {
        m[i] = 0.0f; s[i] = 0.0f; w[i] = 0.0f;
    }
}

// ---------------- pass 1: per-dst feature max of msg ----------------
// msg = relu(x[src]) + eps  > 0, so integer atomicMax on the float bits is
// exact, and m initialized to 0.0f matches the reference's isolated-node fixup.
__global__ void edge_max_kernel(const float* __restrict__ x,
                                const int* __restrict__ src,
                                const int* __restrict__ dst,
                                float* __restrict__ m,
                                int d, int total /* E * d/4 */) {
    const int d4 = d >> 2;
    for (int i = blockIdx.x * blockDim.x + threadIdx.x; i < total;
         i += gridDim.x * blockDim.x) {
        int e = i / d4;
        int f = (i - e * d4) << 2;
        int sn = src[e], dn = dst[e];
        float4 v = *reinterpret_cast<const float4*>(x + (size_t)sn * d + f);
        float4 msg;
        msg.x = fmaxf(v.x, 0.0f) + MSG_EPS;
        msg.y = fmaxf(v.y, 0.0f) + MSG_EPS;
        msg.z = fmaxf(v.z, 0.0f) + MSG_EPS;
        msg.w = fmaxf(v.w, 0.0f) + MSG_EPS;
        int* mp = reinterpret_cast<int*>(m + (size_t)dn * d + f);
        atomicMax(mp + 0, __float_as_int(msg.x));
        atomicMax(mp + 1, __float_as_int(msg.y));
        atomicMax(mp + 2, __float_as_int(msg.z));
        atomicMax(mp + 3, __float_as_int(msg.w));
    }
}

// ---------------- pass 2: ex = exp(msg - m[dst]); accumulate s and msg*ex ----
__global__ void edge_sum_kernel(const float* __restrict__ x,
                                const int* __restrict__ src,
                                const int* __restrict__ dst,
                                const float* __restrict__ m,
                                float* __restrict__ s,
                                float* __restrict__ w,
                                int d, int total /* E * d/4 */) {
    const int d4 = d >> 2;
    for (int i = blockIdx.x * blockDim.x + threadIdx.x; i < total;
         i += gridDim.x * blockDim.x) {
        int e = i / d4;
        int f = (i - e * d4) << 2;
        int sn = src[e], dn = dst[e];
        float4 v = *reinterpret_cast<const float4*>(x + (size_t)sn * d + f);
        float4 mm = *reinterpret_cast<const float4*>(m + (size_t)dn * d + f);
        float mg, ex;
        size_t base = (size_t)dn * d + f;
        mg = fmaxf(v.x, 0.0f) + MSG_EPS; ex = __expf(mg - mm.x);
        atomicAdd(s + base + 0, ex); atomicAdd(w + base + 0, mg * ex);
        mg = fmaxf(v.y, 0.0f) + MSG_EPS; ex = __expf(mg - mm.y);
        atomicAdd(s + base + 1, ex); atomicAdd(w + base + 1, mg * ex);
        mg = fmaxf(v.z, 0.0f) + MSG_EPS; ex = __expf(mg - mm.z);
        atomicAdd(s + base + 2, ex); atomicAdd(w + base + 2, mg * ex);
        mg = fmaxf(v.w, 0.0f) + MSG_EPS; ex = __expf(mg - mm.w);
        atomicAdd(s + base + 3, ex); atomicAdd(w + base + 3, mg * ex);
    }
}

// ---------------- combine: agg = wsum/(s+1e-16) + x (residual) ----------------
// agg may alias the (now dead) m buffer.
__global__ void node_combine_kernel(const float* __restrict__ s,
                                    const float* __restrict__ w,
                                    const float* __restrict__ x,
                                    float* __restrict__ agg, int total) {
    for (int i = blockIdx.x * blockDim.x + threadIdx.x; i < total;
         i += gridDim.x * blockDim.x) {
        agg[i] = w[i] / (s[i] + 1e-16f) + x[i];
    }
}

// ---------------- fp32 WMMA GEMM: H[N x C] = A[N x K] * B[K x C] + bias ------
// One wave per 16x16 tile; V_WMMA_F32_16X16X4_F32, K-step 4.
// A 16x4 layout: lanes 0-15 -> K=0,1 ; lanes 16-31 -> K=2,3 (2 VGPRs).
// B 4x16 / C 16x16 layouts: rows striped across lanes within a VGPR.
__global__ void gemm_wmma_kernel(const float* __restrict__ A,
                                 const float* __restrict__ B,
                                 const float* __restrict__ bias,
                                 float* __restrict__ H,
                                 int K, int C, int tilesN, int relu) {
    int wave = (blockIdx.x * blockDim.x + threadIdx.x) >> 5;
    int lane = threadIdx.x & 31;
    int tm = wave / tilesN;
    int tn = wave - tm * tilesN;
    if (tm >= NNODES / 16) return;          // uniform per wave: EXEC stays full
    int l15 = lane & 15;
    int kh  = lane >> 4;                    // 0 or 1 (K half)
    int row = tm * 16 + l15;
    int col = tn * 16 + l15;
    bool colOK = (col < C);
    int colc = colOK ? col : 0;             // clamped, loads masked via select
    const float* Arow = A + (size_t)row * K + 2 * kh;
    v8f acc = {};
    for (int k0 = 0; k0 < K; k0 += 4) {
        v2f a, b;
        a.x = Arow[k0];
        a.y = Arow[k0 + 1];
        float b0 = B[(size_t)(k0 + 2 * kh) * C + colc];
        float b1 = B[(size_t)(k0 + 2 * kh + 1) * C + colc];
        b.x = colOK ? b0 : 0.0f;
        b.y = colOK ? b1 : 0.0f;
        acc = __builtin_amdgcn_wmma_f32_16x16x4_f32(
            false, a, false, b, (short)0, acc, false, false);
    }
    if (colOK) {
        float bv = bias[col];
        float* Hp = H + col;
#pragma unroll
        for (int i = 0; i < 8; ++i) {
            int r = tm * 16 + kh * 8 + i;   // C/D layout: VGPR i -> M=i (+8)
            float v = acc[i] + bv;
            if (relu) v = fmaxf(v, 0.0f);
            Hp[(size_t)r * C] = v;
        }
    }
}

// ---------------- BatchNorm batch statistics: per-column sum / sumsq ---------
__global__ void bnstats_kernel(const float* __restrict__ H,
                               float* __restrict__ stats, int C) {
    __shared__ float ssum[256];
    __shared__ float ssq[256];
    int c = blockIdx.x;
    int t = threadIdx.x;
    float a = 0.0f, b = 0.0f;
    for (int r = t; r < NNODES; r += 256) {
        float v = H[(size_t)r * C + c];
        a += v; b += v * v;
    }
    ssum[t] = a; ssq[t] = b;
    __syncthreads();
    for (int o = 128; o > 0; o >>= 1) {
        if (t < o) { ssum[t] += ssum[t + o]; ssq[t] += ssq[t + o]; }
        __syncthreads();
    }
    if (t == 0) { stats[c] = ssum[0]; stats[512 + c] = ssq[0]; }
}

// ---------------- BN normalize (train stats) + ReLU, in place ----------------
__global__ void bnrelu_kernel(float* __restrict__ H,
                              const float* __restrict__ stats,
                              const float* __restrict__ g,
                              const float* __restrict__ be,
                              int C, int total) {
    const float invN = 1.0f / (float)NNODES;
    for (int i = blockIdx.x * blockDim.x + threadIdx.x; i < total;
         i += gridDim.x * blockDim.x) {
        int c = i % C;
        float mu  = stats[c] * invN;
        float var = stats[512 + c] * invN - mu * mu;
        float v = (H[i] - mu) * rsqrtf(var + BN_EPS) * g[c] + be[c];
        H[i] = fmaxf(v, 0.0f);
    }
}

// ---------------- final projection 400 -> 2, fused outer ReLU ----------------
__global__ void out_proj_kernel(const float* __restrict__ H,
                                const float* __restrict__ W, /* 400x2 */
                                const float* __restrict__ b,
                                float* __restrict__ out) {
    int i = blockIdx.x * blockDim.x + threadIdx.x;
    if (i >= NNODES * 2) return;
    int n = i >> 1, j = i & 1;
    const float* hp = H + (size_t)n * 400;
    float acc = b[j];
#pragma unroll 4
    for (int k = 0; k < 400; ++k) acc += hp[k] * W[k * 2 + j];
    out[i] = fmaxf(acc, 0.0f);
}

static inline int gblocks(long long n, int bs) {
    long long g = (n + bs - 1) / bs;
    return (int)g;
}

extern "C" void kernel_launch(void* const* d_in, const int* in_sizes, int n_in,
                              void* d_out, int out_size, void* d_ws, size_t ws_size,
                              hipStream_t stream) {
    const float* x     = (const float*)d_in[0];
    const int*   edge  = (const int*)  d_in[1];
    const float* c1_w1 = (const float*)d_in[2];
    const float* c1_b1 = (const float*)d_in[3];
    const float* c1_g  = (const float*)d_in[4];
    const float* c1_be = (const float*)d_in[5];
    const float* c1_w2 = (const float*)d_in[6];
    const float* c1_b2 = (const float*)d_in[7];
    const float* c2_w1 = (const float*)d_in[8];
    const float* c2_b1 = (const float*)d_in[9];
    const float* c2_g  = (const float*)d_in[10];
    const float* c2_be = (const float*)d_in[11];
    const float* c2_w2 = (const float*)d_in[12];
    const float* c2_b2 = (const float*)d_in[13];
    float* out = (float*)d_out;

    const int* src = edge;            // edge_index[0]
    const int* dst = edge + NEDGES;   // edge_index[1]

    // workspace layout (floats)
    float* ws   = (float*)d_ws;
    float* m    = ws;                              // N*200  (also agg, aliased)
    float* s    = m    + (size_t)NNODES * DHID;    // N*200
    float* wsum = s    + (size_t)NNODES * DHID;    // N*200
    float* h1   = wsum + (size_t)NNODES * DHID;    // N*200  layer-1 output
    float* hmid = h1   + (size_t)NNODES * DHID;    // N*400  MLP hidden
    float* stats = hmid + (size_t)NNODES * 2 * DHID; // 1024

    const int BS = 256;

    // ===================== Layer 1 (d = 128) =====================
    {
        const int d = DIN, mid = 2 * DIN;          // 128, 256
        int totN = NNODES * d;
        int totE = NEDGES * (d / 4);
        init3_kernel<<<gblocks(totN, BS), BS, 0, stream>>>(m, s, wsum, totN);
        edge_max_kernel<<<gblocks(totE, BS), BS, 0, stream>>>(x, src, dst, m, d, totE);
        edge_sum_kernel<<<gblocks(totE, BS), BS, 0, stream>>>(x, src, dst, m, s, wsum, d, totE);
        node_combine_kernel<<<gblocks(totN, BS), BS, 0, stream>>>(s, wsum, x, m, totN);
        // h = agg @ w1 + b1   [N x 256]
        int tilesN = (mid + 15) / 16;
        long long waves = (long long)(NNODES / 16) * tilesN;
        gemm_wmma_kernel<<<gblocks(waves * 32, BS), BS, 0, stream>>>(
            m, c1_w1, c1_b1, hmid, d, mid, tilesN, 0);
        bnstats_kernel<<<mid, BS, 0, stream>>>(hmid, stats, mid);
        bnrelu_kernel<<<gblocks((long long)NNODES * mid, BS), BS, 0, stream>>>(
            hmid, stats, c1_g, c1_be, mid, NNODES * mid);
        // h1 = relu(h @ w2 + b2)   [N x 200]  (outer layer ReLU fused)
        tilesN = (DHID + 15) / 16;
        waves = (long long)(NNODES / 16) * tilesN;
        gemm_wmma_kernel<<<gblocks(waves * 32, BS), BS, 0, stream>>>(
            hmid, c1_w2, c1_b2, h1, mid, DHID, tilesN, 1);
    }

    // ===================== Layer 2 (d = 200) =====================
    {
        const int d = DHID, mid = 2 * DHID;        // 200, 400
        int totN = NNODES * d;
        int totE = NEDGES * (d / 4);
        init3_kernel<<<gblocks(totN, BS), BS, 0, stream>>>(m, s, wsum, totN);
        edge_max_kernel<<<gblocks(totE, BS), BS, 0, stream>>>(h1, src, dst, m, d, totE);
        edge_sum_kernel<<<gblocks(totE, BS), BS, 0, stream>>>(h1, src, dst, m, s, wsum, d, totE);
        node_combine_kernel<<<gblocks(totN, BS), BS, 0, stream>>>(s, wsum, h1, m, totN);
        // h = agg @ w1 + b1   [N x 400]
        int tilesN = (mid + 15) / 16;
        long long waves = (long long)(NNODES / 16) * tilesN;
        gemm_wmma_kernel<<<gblocks(waves * 32, BS), BS, 0, stream>>>(
            m, c2_w1, c2_b1, hmid, d, mid, tilesN, 0);
        bnstats_kernel<<<mid, BS, 0, stream>>>(hmid, stats, mid);
        bnrelu_kernel<<<gblocks((long long)NNODES * mid, BS), BS, 0, stream>>>(
            hmid, stats, c2_g, c2_be, mid, NNODES * mid);
        // out = relu(h @ w2 + b2)   [N x 2]
        out_proj_kernel<<<gblocks((long long)NNODES * 2, BS), BS, 0, stream>>>(
            hmid, c2_w2, c2_b2, out);
    }
}